// SoundMvdr_81673098101650
// MI455X (gfx1250) — compile-verified
//
#include <hip/hip_runtime.h>

// MVDR time-recursive beamformer, B=8 C=2 F=513 T=2000, complex64 output.
//
// Single fused kernel using a decoupled-lookback associative scan over T:
//   - 4104 (b,f) chains x 20 chunks of 100 frames -> 82080 lanes (2565 wave32s).
//   - Phase A: stage chunk via CDNA5 async global->LDS B64 copies (ASYNCcnt),
//     compute linear-recurrence aggregate (A = prod ax, b = 4-float Hermitian
//     accumulators) for speech & noise PSDs; publish aggregate (release flag=1).
//     Chunk 0 publishes its inclusive prefix (flag=2) immediately.
//   - Lookback: combine predecessor aggregates until an inclusive prefix is
//     found; publish own inclusive prefix (flag=2).
//   - Phase B: re-stage chunk (L2-hot: re-read happens right after phase A) and
//     emit enhanced output from the exclusive prefix state.
// HBM traffic ~ one input read (164 MB) + output write (66 MB) ~= 20 us floor.
// Deadlock-safe: lanes only wait on lower tids (earlier-dispatched blocks), and
// chain heads publish prefixes unconditionally before anyone spins.

#define B_    8
#define F_    513
#define T_    2000
#define NC_   (B_*F_)          // 4104 chains
#define K_    20               // chunks per chain
#define L_    100              // frames per chunk (chunk 0 == sf 0.9 region)
#define NK_   (NC_*K_)         // 82080 chunk-lanes = 2565 waves exactly
#define TT_   50               // t-tile staged per round
#define LSTR  54               // LDS row stride (floats); 54*s mod 64 distinct for s<32

__device__ __forceinline__ void async_ld_b64(unsigned lds_byte, unsigned voff_byte,
                                             const void* base) {
    // GLOBAL_LOAD_ASYNC_TO_LDS_B64, GVS mode: mem = SADDR + VADDR ; LDS addr in VDST vgpr
    asm volatile("global_load_async_to_lds_b64 %0, %1, %2"
                 :: "v"(lds_byte), "v"(voff_byte), "s"(base) : "memory");
}
__device__ __forceinline__ void wait_async0() {
    asm volatile("s_wait_asynccnt 0" ::: "memory");
}
__device__ __forceinline__ void wait_ds0() {
    asm volatile("s_wait_dscnt 0" ::: "memory");
}

// Stage a 32-stream x TT_ tile of the 5 input streams into LDS (B64 per lane).
__device__ __forceinline__ void stage_tile(const float* __restrict__ re,
                                           const float* __restrict__ im,
                                           const float* __restrict__ mk,
                                           float* sBuf, const unsigned* sOff,
                                           int lane, unsigned t0) {
    unsigned ldsBase = (unsigned)(uintptr_t)sBuf;
    const unsigned a1 = 1u*32*LSTR*4, a2 = 2u*32*LSTR*4, a3 = 3u*32*LSTR*4, a4 = 4u*32*LSTR*4;
    #pragma unroll 5
    for (int j = 0; j < 25; ++j) {                 // 25*32 = 800 pairs = 32 rows * 25 pairs
        int g  = j*32 + lane;
        int s  = g / 25;                           // source stream (lane that owns it)
        int p2 = g - s*25;                         // float-pair index within row
        unsigned rowOff = ldsBase + (unsigned)((s*LSTR + 2*p2)*4);
        unsigned e0 = (sOff[s]    + t0 + (unsigned)(2*p2)) * 4u;  // channel 0
        unsigned e1 = (sOff[32+s] + t0 + (unsigned)(2*p2)) * 4u;  // channel 1
        unsigned em = (sOff[64+s] + t0 + (unsigned)(2*p2)) * 4u;  // mask / out
        async_ld_b64(rowOff      , e0, re);
        async_ld_b64(rowOff + a1 , e0, im);
        async_ld_b64(rowOff + a2 , e1, re);
        async_ld_b64(rowOff + a3 , e1, im);
        async_ld_b64(rowOff + a4 , em, mk);
    }
}

__global__ void mvdr_init(unsigned* __restrict__ flag) {
    const int i = blockIdx.x*blockDim.x + threadIdx.x;
    if (i < NK_) flag[i] = 0u;
}

__global__ __launch_bounds__(32)
void mvdr_fused(const float* __restrict__ re, const float* __restrict__ im,
                const float* __restrict__ mk,
                float* __restrict__ agg,        // 10 x NK_  (A,bS[4] | A,bN[4])
                float* __restrict__ inc,        //  8 x NK_  inclusive prefixes
                unsigned* __restrict__ flag,    // NK_ : 0=none 1=aggregate 2=inclusive
                float2* __restrict__ out) {
    __shared__ float    sBuf[5*32*LSTR];
    __shared__ unsigned sOff[3*32];
    const int lane  = threadIdx.x;
    const int tid   = blockIdx.x*32 + lane;
    const int chain = tid / K_;
    const int chunk = tid - chain*K_;
    const int b = chain / F_;
    const int f = chain - b*F_;
    const unsigned obf = (unsigned)((b*F_ + f)*T_ + chunk*L_);   // mask AND output offset
    const unsigned oc0 = (unsigned)(((b*2 + 0)*F_ + f)*T_ + chunk*L_);
    const unsigned oc1 = oc0 + (unsigned)(F_*T_);
    sOff[lane] = oc0; sOff[32+lane] = oc1; sOff[64+lane] = obf;
    __syncthreads();

    const float sf  = (chunk == 0) ? 0.9f : 0.98f;
    const float osf = 1.0f - sf;

    const float* sR0 = sBuf + 0*32*LSTR + lane*LSTR;
    const float* sI0 = sBuf + 1*32*LSTR + lane*LSTR;
    const float* sR1 = sBuf + 2*32*LSTR + lane*LSTR;
    const float* sI1 = sBuf + 3*32*LSTR + lane*LSTR;
    const float* sM  = sBuf + 4*32*LSTR + lane*LSTR;

    // ---------------- Phase A: chunk-local aggregate -------------------------
    float Ax = 1.f, An = 1.f;
    float bs0=0,bs1=0,bs2=0,bs3=0, bn0=0,bn1=0,bn2=0,bn3=0;
    for (int tile = 0; tile < L_/TT_; ++tile) {
        wait_ds0();                                   // LDS reads done before overwrite
        stage_tile(re, im, mk, sBuf, sOff, lane, (unsigned)(tile*TT_));
        wait_async0();                                // async copies landed in LDS
        for (int p = 0; p < TT_; ++p) {
            float r0 = sR0[p], i0 = sI0[p], r1 = sR1[p], i1 = sI1[p], m = sM[p];
            float p00 = r0*r0 + i0*i0;                // psd = spec (x) conj(spec)
            float p11 = r1*r1 + i1*i1;
            float por = r0*r1 + i0*i1;
            float poi = i0*r1 - r0*i1;
            float wm  = m*osf;                        // 1-ax
            float wn  = osf - wm;                     // 1-an
            float ax  = 1.f - wm;
            float an  = 1.f - wn;
            Ax *= ax;  An *= an;
            bs0 = ax*bs0 + wm*p00;  bs1 = ax*bs1 + wm*p11;
            bs2 = ax*bs2 + wm*por;  bs3 = ax*bs3 + wm*poi;
            bn0 = an*bn0 + wn*p00;  bn1 = an*bn1 + wn*p11;
            bn2 = an*bn2 + wn*por;  bn3 = an*bn3 + wn*poi;
        }
    }

    // ---------------- Publish + decoupled lookback ---------------------------
    float x0=0,x1=0,x2=0,x3=0, y0=0,y1=0,y2=0,y3=0;   // exclusive prefix (S,N)
    if (chunk != 0) {
        // publish aggregate (flag=1, release)
        agg[0*NK_+tid]=Ax;  agg[1*NK_+tid]=bs0; agg[2*NK_+tid]=bs1;
        agg[3*NK_+tid]=bs2; agg[4*NK_+tid]=bs3;
        agg[5*NK_+tid]=An;  agg[6*NK_+tid]=bn0; agg[7*NK_+tid]=bn1;
        agg[8*NK_+tid]=bn2; agg[9*NK_+tid]=bn3;
        __hip_atomic_store(&flag[tid], 1u, __ATOMIC_RELEASE, __HIP_MEMORY_SCOPE_AGENT);

        // lookback: carry transform (cAx,cbS | cAn,cbN), identity to start
        float cAx=1.f, cb0=0,cb1=0,cb2=0,cb3=0;
        float cAn=1.f, cc0=0,cc1=0,cc2=0,cc3=0;
        int j = tid - 1;
        for (;;) {
            unsigned fl;
            while ((fl = __hip_atomic_load(&flag[j], __ATOMIC_ACQUIRE,
                                           __HIP_MEMORY_SCOPE_AGENT)) == 0u)
                __builtin_amdgcn_s_sleep(1);
            if (fl == 2u) {                            // inclusive prefix found
                float pS0=inc[0*NK_+j], pS1=inc[1*NK_+j], pS2=inc[2*NK_+j], pS3=inc[3*NK_+j];
                float pN0=inc[4*NK_+j], pN1=inc[5*NK_+j], pN2=inc[6*NK_+j], pN3=inc[7*NK_+j];
                x0 = cAx*pS0 + cb0;  x1 = cAx*pS1 + cb1;
                x2 = cAx*pS2 + cb2;  x3 = cAx*pS3 + cb3;
                y0 = cAn*pN0 + cc0;  y1 = cAn*pN1 + cc1;
                y2 = cAn*pN2 + cc2;  y3 = cAn*pN3 + cc3;
                break;
            }
            // combine: carry = carry o agg_j
            float jAx=agg[0*NK_+j], jAn=agg[5*NK_+j];
            cb0 = cAx*agg[1*NK_+j] + cb0;  cb1 = cAx*agg[2*NK_+j] + cb1;
            cb2 = cAx*agg[3*NK_+j] + cb2;  cb3 = cAx*agg[4*NK_+j] + cb3;
            cc0 = cAn*agg[6*NK_+j] + cc0;  cc1 = cAn*agg[7*NK_+j] + cc1;
            cc2 = cAn*agg[8*NK_+j] + cc2;  cc3 = cAn*agg[9*NK_+j] + cc3;
            cAx *= jAx;  cAn *= jAn;
            --j;                                       // bounded: stops at chain's chunk 0
        }
    }
    // publish inclusive prefix for successor (last chunk of a chain has none)
    if (chunk != K_-1) {
        inc[0*NK_+tid] = Ax*x0 + bs0;  inc[1*NK_+tid] = Ax*x1 + bs1;
        inc[2*NK_+tid] = Ax*x2 + bs2;  inc[3*NK_+tid] = Ax*x3 + bs3;
        inc[4*NK_+tid] = An*y0 + bn0;  inc[5*NK_+tid] = An*y1 + bn1;
        inc[6*NK_+tid] = An*y2 + bn2;  inc[7*NK_+tid] = An*y3 + bn3;
        __hip_atomic_store(&flag[tid], 2u, __ATOMIC_RELEASE, __HIP_MEMORY_SCOPE_AGENT);
    }

    // ---------------- Phase B: re-walk chunk from prefix, emit output --------
    float s0=x0,s1=x1,s2=x2,s3=x3, n0=y0,n1=y1,n2=y2,n3=y3;
    for (int tile = 0; tile < L_/TT_; ++tile) {
        const unsigned t0 = (unsigned)(tile*TT_);
        wait_ds0();
        stage_tile(re, im, mk, sBuf, sOff, lane, t0);  // L2-hot re-read
        wait_async0();
        for (int p = 0; p < TT_; ++p) {
            float r0 = sR0[p], i0 = sI0[p], r1 = sR1[p], i1 = sI1[p], m = sM[p];
            float p00 = r0*r0 + i0*i0;
            float p11 = r1*r1 + i1*i1;
            float por = r0*r1 + i0*i1;
            float poi = i0*r1 - r0*i1;
            float wm  = m*osf, wn = osf - wm;
            float ax  = 1.f - wm, an = 1.f - wn;
            s0 = ax*s0 + wm*p00;  s1 = ax*s1 + wm*p11;
            s2 = ax*s2 + wm*por;  s3 = ax*s3 + wm*poi;
            n0 = an*n0 + wn*p00;  n1 = an*n1 + wn*p11;
            n2 = an*n2 + wn*por;  n3 = an*n3 + wn*poi;
            // numerator = S * adj(N); column 0 and (exactly real) trace
            float cross = s2*n2 + s3*n3;
            float w0r = s0*n1 - cross;          // num[0][0].re
            float w0i = s2*n3 - s3*n2;          // num[0][0].im
            float w1r = s2*n1 - s1*n2;          // num[1][0].re
            float w1i = s1*n3 - s3*n1;          // num[1][0].im
            float tr  = s0*n1 + s1*n0 - 2.f*cross + 1e-16f;
            float inv = 1.f / tr;
            w0r *= inv; w0i *= inv; w1r *= inv; w1i *= inv;
            // enhanced = conj(w0)*spec0 + conj(w1)*spec1
            float er = w0r*r0 + w0i*i0 + w1r*r1 + w1i*i1;
            float ei = w0r*i0 - w0i*r0 + w1r*i1 - w1i*r1;
            out[obf + t0 + (unsigned)p] = make_float2(er, ei);
        }
    }
}

extern "C" void kernel_launch(void* const* d_in, const int* in_sizes, int n_in,
                              void* d_out, int out_size, void* d_ws, size_t ws_size,
                              hipStream_t stream) {
    const float* re = (const float*)d_in[0];   // (B,C,F,T) f32
    const float* im = (const float*)d_in[1];   // (B,C,F,T) f32
    const float* mk = (const float*)d_in[2];   // (B,1,F,T) f32
    float2* out = (float2*)d_out;              // (B,F,T) complex64 interleaved

    float*    agg  = (float*)d_ws;                 // 10 * NK_ floats
    float*    inc  = agg + 10*NK_;                 //  8 * NK_ floats
    unsigned* flag = (unsigned*)(inc + 8*NK_);     //  NK_ u32   (total ~6.2 MB)

    mvdr_init <<<(NK_ + 255)/256, 256, 0, stream>>>(flag);
    mvdr_fused<<<NK_/32, 32, 0, stream>>>(re, im, mk, agg, inc, flag, out);
    (void)in_sizes; (void)n_in; (void)out_size; (void)ws_size;
}